// P2PSigned_50740743635776
// MI455X (gfx1250) — compile-verified
//
#include <hip/hip_runtime.h>
#include <hip/hip_bf16.h>
#include <cmath>

typedef float v2f __attribute__((ext_vector_type(2)));
typedef float v8f __attribute__((ext_vector_type(8)));

#define WAVES_PER_BLOCK 8
#define PTS_PER_WAVE    16
#define PTS_PER_BLOCK   (WAVES_PER_BLOCK * PTS_PER_WAVE)

// One wave owns 16 query points and streams the database in 16-point tiles.
// Per tile, a single V_WMMA_F32_16X16X4_F32 produces a row-order-equivalent
// squared-distance tile:
//   A row m = [-2x0, -2x1, -2x2, 1]      (loop-invariant, in 2 VGPRs)
//   B col q = [ y0,   y1,   y2,  y_q^2]  (K=3 slot carries |y|^2)
//   C       = inline 0
//   D[m][q] = |y_q|^2 - 2 x.y  ==  d2[m][q] - |x_m|^2
// The -|x_m|^2 bias is constant per row, so every argmin comparison (in-loop
// and in the cross-lane butterfly) is unaffected; the true distance is
// recomputed from raw coordinates for the final signed output.
__global__ __launch_bounds__(256) void nn_signed_kernel(
    const float* __restrict__ qry,   // (N, Pq, 3)  query points
    const float* __restrict__ db,    // (N, Pdb, 3) database points
    const float* __restrict__ dbn,   // (N, Pdb, 3) database normals
    float* __restrict__ out_signed,  // (N, Pq)
    float* __restrict__ out_idx,     // (N, Pq), index stored as float
    int Pq, int Pdb)
{
  const int n    = blockIdx.y;
  const int lane = threadIdx.x & 31;
  const int wave = threadIdx.x >> 5;
  const int l15  = lane & 15;
  const int hi   = lane >> 4;          // which 16-lane half of the wave

  const float* __restrict__ Q  = qry + (size_t)n * Pq  * 3;
  const float* __restrict__ Bp = db  + (size_t)n * Pdb * 3;
  const float* __restrict__ BN = dbn + (size_t)n * Pdb * 3;

  const int pbase = blockIdx.x * PTS_PER_BLOCK + wave * PTS_PER_WAVE;

  // Owned query point for this lane (row M = l15 of the A matrix).
  const int p = pbase + l15;
  const float x0 = Q[3 * p + 0];
  const float x1 = Q[3 * p + 1];
  const float x2 = Q[3 * p + 2];

  // A-matrix (16x4 f32): lanes 0-15 hold K=0,1 in VGPR0,1; lanes 16-31 K=2,3.
  v2f A;
  A.x = hi ? (-2.0f * x2) : (-2.0f * x0);
  A.y = hi ? 1.0f         : (-2.0f * x1);

  float minv[8];
  int   mini[8];
#pragma unroll
  for (int i = 0; i < 8; ++i) { minv[i] = __builtin_inff(); mini[i] = 0; }

  for (int jb = 0; jb < Pdb; jb += 16) {
    const int j = jb + l15;            // database column this lane feeds
    const float w0 = Bp[3 * j + 0];
    const float w1 = Bp[3 * j + 1];
    const float w2 = Bp[3 * j + 2];
    const float ysq = w0 * w0 + w1 * w1 + w2 * w2;

    // B-matrix (4x16 f32): lanes 0-15 hold K=0,1; lanes 16-31 K=2,3
    // (row K=3 carries |y_q|^2, multiplied by A's constant-1 column).
    v2f Bm;
    Bm.x = hi ? w2  : w0;
    Bm.y = hi ? ysq : w1;

    v8f C = {};  // inline zero accumulator

    v8f D = __builtin_amdgcn_wmma_f32_16x16x4_f32(
        /*neg_a=*/false, A, /*neg_b=*/false, Bm,
        /*c_mod=*/(short)0, C, /*reuse_a=*/false, /*reuse_b=*/false);

#pragma unroll
    for (int i = 0; i < 8; ++i) {
      const float dv = D[i];
      const bool lt = dv < minv[i];
      minv[i] = lt ? dv : minv[i];
      mini[i] = lt ? j  : mini[i];
    }
  }

  // Butterfly argmin across the 16 lanes sharing each row (masks stay in-half).
#pragma unroll
  for (int m = 8; m >= 1; m >>= 1) {
#pragma unroll
    for (int i = 0; i < 8; ++i) {
      const float ov = __shfl_xor(minv[i], m, 32);
      const int   oi = __shfl_xor(mini[i], m, 32);
      const bool take = (ov < minv[i]) || ((ov == minv[i]) && (oi < mini[i]));
      minv[i] = take ? ov : minv[i];
      mini[i] = take ? oi : mini[i];
    }
  }

  // Lane 0 of each half holds results for rows M = hi*8 + i.
  if (l15 == 0) {
#pragma unroll
    for (int i = 0; i < 8; ++i) {
      const int pp = pbase + hi * 8 + i;
      const int j  = mini[i];
      const float d0 = Q[3 * pp + 0] - Bp[3 * j + 0];
      const float d1 = Q[3 * pp + 1] - Bp[3 * j + 1];
      const float d2 = Q[3 * pp + 2] - Bp[3 * j + 2];
      const float nrm = sqrtf(d0 * d0 + d1 * d1 + d2 * d2);
      const float dt  = BN[3 * j + 0] * d0 + BN[3 * j + 1] * d1 + BN[3 * j + 2] * d2;
      const float sgn = dt > 0.0f ? 1.0f : (dt < 0.0f ? -1.0f : 0.0f);
      out_signed[(size_t)n * Pq + pp] = nrm * sgn;
      out_idx   [(size_t)n * Pq + pp] = (float)j;
    }
  }
}

extern "C" void kernel_launch(void* const* d_in, const int* in_sizes, int n_in,
                              void* d_out, int out_size, void* d_ws, size_t ws_size,
                              hipStream_t stream) {
  (void)in_sizes; (void)n_in; (void)out_size; (void)d_ws; (void)ws_size;

  const int N = 4, P1 = 8192, P2 = 8192;

  const float* x  = (const float*)d_in[0];
  const float* y  = (const float*)d_in[1];
  const float* xn = (const float*)d_in[2];
  const float* yn = (const float*)d_in[3];

  float* out        = (float*)d_out;
  float* y2x_signed = out;                                   // (N, P2)
  float* x2y_signed = out + (size_t)N * P2;                  // (N, P1)
  float* yidx       = x2y_signed + (size_t)N * P1;           // (N, P2)
  float* xidx       = yidx + (size_t)N * P2;                 // (N, P1)

  dim3 block(32 * WAVES_PER_BLOCK);

  // Pass 1: queries = x, database = y (+y_normals) -> x2y_signed, xidx_near
  nn_signed_kernel<<<dim3(P1 / PTS_PER_BLOCK, N), block, 0, stream>>>(
      x, y, yn, x2y_signed, xidx, P1, P2);

  // Pass 2: queries = y, database = x (+x_normals) -> y2x_signed, yidx_near
  nn_signed_kernel<<<dim3(P2 / PTS_PER_BLOCK, N), block, 0, stream>>>(
      y, x, xn, y2x_signed, yidx, P2, P1);
}